// IoULoss_8400956031598
// MI455X (gfx1250) — compile-verified
//
#include <hip/hip_runtime.h>
#include <stdint.h>

// Problem constants from the reference
#define B_  128
#define N_  256
#define H_  256
#define W_  256
#define HW_ (H_ * W_)

typedef float v2f __attribute__((ext_vector_type(2)));
typedef float v8f __attribute__((ext_vector_type(8)));

__device__ __forceinline__ float readlane_f(float x, int lane) {
    return __builtin_bit_cast(float,
        __builtin_amdgcn_readlane(__builtin_bit_cast(int, x), lane));
}

// Full-precision wave32 sum via V_WMMA_F32_16X16X4_F32.
// A (16x4): lanes 0-15 -> (M=lane, K=0|1), lanes 16-31 -> (M=lane-16, K=2|3).
// With a.y = 0 and B = ones: D[m,n] = v[m] + v[m+16] for all n.
// D layout: lanes 0-15 hold rows M=0..7 in d[0..7] (col n=lane),
//           lanes 16-31 hold rows M=8..15.
// => sum(d) @lane0 = sum v[0..7] + v[16..23]; @lane16 = sum v[8..15] + v[24..31].
__device__ __forceinline__ float wave_sum32_wmma(float v) {
    v2f a;  a[0] = v;    a[1] = 0.0f;
    v2f bm; bm[0] = 1.0f; bm[1] = 1.0f;   // all-ones B (layout-independent)
    v8f c = {};
    v8f d = __builtin_amdgcn_wmma_f32_16x16x4_f32(
        /*neg_a=*/false, a, /*neg_b=*/false, bm,
        /*c_mod=*/(short)0, c, /*reuse_a=*/false, /*reuse_b=*/false);
    float s = ((d[0] + d[1]) + (d[2] + d[3])) + ((d[4] + d[5]) + (d[6] + d[7]));
    return readlane_f(s, 0) + readlane_f(s, 16);
}

__global__ void iou_zero_kernel(float* __restrict__ acc) {
    acc[0] = 0.0f;  // loss sum
    acc[1] = 0.0f;  // mask count
}

__global__ __launch_bounds__(256) void iou_main_kernel(
    const float*   __restrict__ out,     // [B,2,H,W]
    const float2*  __restrict__ target,  // [B,N,2]
    const int*     __restrict__ ind,     // [B,N]
    const uint8_t* __restrict__ mask,    // [B,N] (bool)
    float*         __restrict__ acc)     // [2] accumulators in d_ws
{
    const int i = blockIdx.x * 256 + threadIdx.x;   // i < B_*N_ (exact grid)
    const int b = i >> 8;                           // N_ == 256

    const int    idx = ind[i];
    const float2 t   = target[i];

    const float* plane = out + (size_t)b * (2 * HW_);
    const float dx = plane[idx];          // channel 0
    const float dy = plane[HW_ + idx];    // channel 1

    const float xmin = (float)(idx & (W_ - 1));     // ind % W
    const float ymin = (float)(idx >> 8);           // ind / W

    // torch .int() truncation toward zero
    const float xg = truncf(xmin - t.x * 0.5f);
    const float yg = truncf(ymin - t.y * 0.5f);
    const float xd = truncf(xmin - dx * 0.5f);
    const float yd = truncf(ymin - dy * 0.5f);

    // boxes (cx, cy, w, h) -> corners
    const float g_x1 = xg - t.x * 0.5f, g_x2 = xg + t.x * 0.5f;
    const float g_y1 = yg - t.y * 0.5f, g_y2 = yg + t.y * 0.5f;
    const float d_x1 = xd - dx  * 0.5f, d_x2 = xd + dx  * 0.5f;
    const float d_y1 = yd - dy  * 0.5f, d_y2 = yd + dy  * 0.5f;

    const float iw = fmaxf(fminf(g_x2, d_x2) - fmaxf(g_x1, d_x1), 0.0f);
    const float ih = fmaxf(fminf(g_y2, d_y2) - fmaxf(g_y1, d_y1), 0.0f);
    const float inter = iw * ih;

    const float w1 = g_x2 - g_x1, h1 = g_y2 - g_y1;
    const float w2 = d_x2 - d_x1, h2 = d_y2 - d_y1;
    const float uni = w1 * h1 + 1e-16f + w2 * h2 - inter;
    const float iou = inter / uni;

    const bool  m    = (mask[i] != 0);
    const float loss = m ? (1.0f - iou) : 0.0f;  // select, no divergence:
    const float cnt  = m ? 1.0f : 0.0f;          // EXEC stays all-1s for WMMA

    // wave32 reductions via fp32 WMMA (uniform result across the wave)
    const float lossSum = wave_sum32_wmma(loss);
    const float cntSum  = wave_sum32_wmma(cnt);

    if ((threadIdx.x & 31) == 0) {
        atomicAdd(&acc[0], lossSum);
        atomicAdd(&acc[1], cntSum);
    }
}

__global__ void iou_final_kernel(const float* __restrict__ acc,
                                 float* __restrict__ out) {
    out[0] = acc[0] / (4.0f * acc[1] + 0.0001f);
}

extern "C" void kernel_launch(void* const* d_in, const int* in_sizes, int n_in,
                              void* d_out, int out_size, void* d_ws, size_t ws_size,
                              hipStream_t stream) {
    (void)in_sizes; (void)n_in; (void)out_size; (void)ws_size;

    const float*   out_map = (const float*)  d_in[0];  // [B,2,H,W] f32
    const float2*  target  = (const float2*) d_in[1];  // [B,N,2]   f32
    const int*     ind     = (const int*)    d_in[2];  // [B,N]     i32
    const uint8_t* mask    = (const uint8_t*)d_in[3];  // [B,N]     bool
    float*         result  = (float*)d_out;            // scalar f32
    float*         acc     = (float*)d_ws;             // 2 floats of scratch

    iou_zero_kernel<<<1, 1, 0, stream>>>(acc);

    const int total  = B_ * N_;            // 32768
    const int blocks = total / 256;        // 128 blocks x 8 waves
    iou_main_kernel<<<blocks, 256, 0, stream>>>(out_map, target, ind, mask, acc);

    iou_final_kernel<<<1, 1, 0, stream>>>(acc, result);
}